// Net_3layers_57767310131232
// MI455X (gfx1250) — compile-verified
//
#include <hip/hip_runtime.h>
#include <hip/hip_bf16.h>
#include <math.h>

// ---------------------------------------------------------------------------
// GCN 3-layer pipeline for MI455X (gfx1250), f32 WMMA path.
// ---------------------------------------------------------------------------

typedef float v2f __attribute__((ext_vector_type(2)));
typedef float v8f __attribute__((ext_vector_type(8)));

static constexpr int N_NODES   = 50000;
static constexpr int N_EDGES   = 800000;
static constexpr int NFEAT     = 256;
static constexpr int HIDDEN    = 128;
static constexpr int NCLASSES  = 40;
static constexpr int NCLS_PAD  = 48;   // padded to 3 WMMA n-tiles

// ----------------------------- WMMA GEMM -----------------------------------
// C[M,N] = op(A[M,K]) @ B[K,N], row-major, N/K compile-time. Each wave computes
// a 16 x (16*NT) output slab: one shared A fragment per k-step feeds NT
// independent V_WMMA_F32_16X16X4_F32 issues (exact f32 math on CDNA5).
// Depth-1 software pipeline: k+4 fragments are in flight while k's WMMAs run.
// op(A) = relu(A) when RELU.
template<int NT, bool RELU, int N, int K>
__global__ void gemm_wmma_f32(const float* __restrict__ A,
                              const float* __restrict__ B,
                              float* __restrict__ C,
                              int M)
{
    constexpr int ngroups = N / (16 * NT);
    const int wavesPerBlock = blockDim.x >> 5;
    const int wave  = blockIdx.x * wavesPerBlock + (threadIdx.x >> 5);
    const int total = (M >> 4) * ngroups;
    if (wave >= total) return;               // wave-uniform exit: EXEC stays all-1s

    const int tm   = wave / ngroups;
    const int tg   = wave - tm * ngroups;
    const int lane = threadIdx.x & 31;
    const int half = lane >> 4;              // 0: K pair {k,k+1}, 1: {k+2,k+3}
    const int lr   = lane & 15;
    const int m0   = tm << 4;
    const int n0   = tg * (16 * NT);

    v8f acc[NT] = {};
    const float* __restrict__ aP = A + (size_t)(m0 + lr) * K + (half << 1);
    const float* __restrict__ bP = B + (half << 1) * N + n0 + lr;

    // prologue: fragments for k = 0
    v2f af = *(const v2f*)aP;
    v2f bf[NT];
#pragma unroll
    for (int t = 0; t < NT; ++t) {
        bf[t].x = bP[16 * t];
        bf[t].y = bP[16 * t + N];
    }

#pragma unroll 2
    for (int k = 0; k < K - 4; k += 4) {
        // prefetch fragments for k+4 (always valid: loop stops at K-8)
        v2f afn = *(const v2f*)(aP + 4);
        v2f bfn[NT];
        const float* __restrict__ bPn = bP + 4 * N;
#pragma unroll
        for (int t = 0; t < NT; ++t) {
            bfn[t].x = bPn[16 * t];
            bfn[t].y = bPn[16 * t + N];
        }
        // consume fragments for k
        v2f au = af;
        if (RELU) { au.x = fmaxf(au.x, 0.0f); au.y = fmaxf(au.y, 0.0f); }
#pragma unroll
        for (int t = 0; t < NT; ++t)
            acc[t] = __builtin_amdgcn_wmma_f32_16x16x4_f32(
                         false, au, false, bf[t], (short)0, acc[t], false, false);
        // rotate
        af = afn;
#pragma unroll
        for (int t = 0; t < NT; ++t) bf[t] = bfn[t];
        aP += 4;
        bP += 4 * N;
    }

    // epilogue: final k-step
    {
        v2f au = af;
        if (RELU) { au.x = fmaxf(au.x, 0.0f); au.y = fmaxf(au.y, 0.0f); }
#pragma unroll
        for (int t = 0; t < NT; ++t)
            acc[t] = __builtin_amdgcn_wmma_f32_16x16x4_f32(
                         false, au, false, bf[t], (short)0, acc[t], false, false);
    }

    // C/D layout: VGPR r holds row m0+r (lanes 0-15) / m0+8+r (lanes 16-31)
#pragma unroll
    for (int t = 0; t < NT; ++t) {
        float* __restrict__ Crow =
            C + (size_t)(m0 + (half << 3)) * N + n0 + 16 * t + lr;
#pragma unroll
        for (int r = 0; r < 8; ++r)
            Crow[(size_t)r * N] = acc[t][r];
    }
}

// ------------------------ bias broadcast init ------------------------------
__global__ void bias_init(float* __restrict__ agg, const float* __restrict__ b,
                          int n, int d)
{
    size_t i   = (size_t)blockIdx.x * blockDim.x + threadIdx.x;
    size_t tot = (size_t)n * d;
    if (i < tot) agg[i] = b[i % d];
}

// ---------------------- edge scatter-add (SpMM) ----------------------------
// One wave per edge; lane L handles float4 at column 4*L.
// agg[dst] += H[src] * w   via global_atomic_add_f32 (L2-resident buffer).
__global__ void scatter_add(const float* __restrict__ H,
                            const int*   __restrict__ src,
                            const int*   __restrict__ dst,
                            const float* __restrict__ w,
                            float* __restrict__ agg,
                            int E, int ldh, int d)
{
    const int e = blockIdx.x * (blockDim.x >> 5) + (threadIdx.x >> 5);
    if (e >= E) return;
    const int lane = threadIdx.x & 31;
    const int s  = src[e];
    const int t  = dst[e];
    const float we = w[e];
    const int col = lane << 2;
    if (col < d) {
        const float4 v = *(const float4*)(H + (size_t)s * ldh + col);
        float* ap = agg + (size_t)t * d + col;
        atomicAdd(ap + 0, v.x * we);
        atomicAdd(ap + 1, v.y * we);
        atomicAdd(ap + 2, v.z * we);
        atomicAdd(ap + 3, v.w * we);
    }
}

// ------------------------- pad W3 40 -> 48 cols ----------------------------
__global__ void pad_w3(const float* __restrict__ W3, float* __restrict__ W3p)
{
    int i = blockIdx.x * blockDim.x + threadIdx.x;
    if (i < HIDDEN * NCLS_PAD) {
        int r = i / NCLS_PAD, c = i - r * NCLS_PAD;
        W3p[i] = (c < NCLASSES) ? W3[r * NCLASSES + c] : 0.0f;
    }
}

// ------------------------ log_softmax over 40 ------------------------------
// One wave per row; lanes 0..31 hold cols 0..31, lanes 0..7 also cols 32..39.
__global__ void log_softmax40(const float* __restrict__ X,
                              float* __restrict__ Y, int n)
{
    const int row = blockIdx.x * (blockDim.x >> 5) + (threadIdx.x >> 5);
    if (row >= n) return;
    const int lane = threadIdx.x & 31;
    const float* xr = X + (size_t)row * NCLASSES;

    float v0 = xr[lane];
    float v1 = (lane < 8) ? xr[32 + lane] : -INFINITY;

    float m = fmaxf(v0, v1);
#pragma unroll
    for (int off = 16; off > 0; off >>= 1)
        m = fmaxf(m, __shfl_xor(m, off, 32));

    float s = expf(v0 - m) + ((lane < 8) ? expf(v1 - m) : 0.0f);
#pragma unroll
    for (int off = 16; off > 0; off >>= 1)
        s += __shfl_xor(s, off, 32);

    const float ls = m + logf(s);
    float* yr = Y + (size_t)row * NCLASSES;
    yr[lane] = v0 - ls;
    if (lane < 8) yr[32 + lane] = v1 - ls;
}

// ---------------------------------------------------------------------------
extern "C" void kernel_launch(void* const* d_in, const int* in_sizes, int n_in,
                              void* d_out, int out_size, void* d_ws, size_t ws_size,
                              hipStream_t stream)
{
    const float* x    = (const float*)d_in[0];
    const int*   esrc = (const int*)  d_in[1];
    const int*   edst = (const int*)  d_in[2];
    const float* ew   = (const float*)d_in[3];
    const float* W1   = (const float*)d_in[4];
    const float* b1   = (const float*)d_in[5];
    const float* W2   = (const float*)d_in[6];
    const float* b2   = (const float*)d_in[7];
    const float* W3   = (const float*)d_in[8];
    const float* b3   = (const float*)d_in[9];
    float* out = (float*)d_out;

    // Workspace: two ping-pong node buffers + padded W3
    const size_t bufBytes = (size_t)N_NODES * HIDDEN * sizeof(float); // 25.6 MB
    float* bufA  = (float*)d_ws;                                   // H1 / H2 / H3
    float* bufB  = (float*)((char*)d_ws + bufBytes);               // AGG1 / AGG2 / AGG3
    float* W3pad = (float*)((char*)d_ws + 2 * bufBytes);           // 128x48

    const int tpb = 256;                 // 8 waves per block
    const int wavesPerBlock = tpb / 32;

    auto gemmBlocks = [&](int M, int N, int NT) {
        int waves = (M / 16) * (N / (16 * NT));
        return (waves + wavesPerBlock - 1) / wavesPerBlock;
    };
    const int edgeBlocks = (N_EDGES + wavesPerBlock - 1) / wavesPerBlock;

    // ---- Layer 1: H1 = X @ W1 ; AGG1 = scatter(H1) + b1 ----
    gemm_wmma_f32<4, false, HIDDEN, NFEAT>
        <<<gemmBlocks(N_NODES, HIDDEN, 4), tpb, 0, stream>>>(x, W1, bufA, N_NODES);
    bias_init<<<(int)(((size_t)N_NODES * HIDDEN + tpb - 1) / tpb), tpb, 0, stream>>>(
        bufB, b1, N_NODES, HIDDEN);
    scatter_add<<<edgeBlocks, tpb, 0, stream>>>(
        bufA, esrc, edst, ew, bufB, N_EDGES, HIDDEN, HIDDEN);

    // ---- Layer 2: H2 = relu(AGG1) @ W2 ; AGG2 = scatter(H2) + b2 ----
    gemm_wmma_f32<4, true, HIDDEN, HIDDEN>
        <<<gemmBlocks(N_NODES, HIDDEN, 4), tpb, 0, stream>>>(bufB, W2, bufA, N_NODES);
    bias_init<<<(int)(((size_t)N_NODES * HIDDEN + tpb - 1) / tpb), tpb, 0, stream>>>(
        bufB, b2, N_NODES, HIDDEN);
    scatter_add<<<edgeBlocks, tpb, 0, stream>>>(
        bufA, esrc, edst, ew, bufB, N_EDGES, HIDDEN, HIDDEN);

    // ---- Layer 3: H3 = relu(AGG2) @ pad(W3) ; AGG3 = scatter(H3) + b3 ----
    pad_w3<<<(HIDDEN * NCLS_PAD + tpb - 1) / tpb, tpb, 0, stream>>>(W3, W3pad);
    gemm_wmma_f32<3, true, NCLS_PAD, HIDDEN>
        <<<gemmBlocks(N_NODES, NCLS_PAD, 3), tpb, 0, stream>>>(bufB, W3pad, bufA, N_NODES);
    bias_init<<<(int)(((size_t)N_NODES * NCLASSES + tpb - 1) / tpb), tpb, 0, stream>>>(
        bufB, b3, N_NODES, NCLASSES);
    scatter_add<<<edgeBlocks, tpb, 0, stream>>>(
        bufA, esrc, edst, ew, bufB, N_EDGES, NCLS_PAD, NCLASSES);

    // ---- log_softmax ----
    const int lsBlocks = (N_NODES + wavesPerBlock - 1) / wavesPerBlock;
    log_softmax40<<<lsBlocks, tpb, 0, stream>>>(bufB, out, N_NODES);
}